// NeRFRenderer_81329500717182
// MI455X (gfx1250) — compile-verified
//
#include <hip/hip_runtime.h>
#include <math.h>

// NeRF occupancy-grid renderer for MI455X (gfx1250, wave32).
// Pass 1: density MLP via V_WMMA_F32_16X16X4_F32 (bias folded into K=4).
// Pass 2: per-ray segmented cumsum / transmittance / accumulation.

#define NERF_HIDDEN 64

typedef __attribute__((ext_vector_type(2))) float v2f;
typedef __attribute__((ext_vector_type(8))) float v8f;

__device__ __forceinline__ float softplus_stable(float x) {
    // log(1+exp(x)) = max(x,0) + log1p(exp(-|x|))
    return fmaxf(x, 0.0f) + log1pf(expf(-fabsf(x)));
}

// One wave (32 lanes) handles one tile of 16 samples.
// A (16x4 f32): lanes 0-15 hold K=0,1 (pos.x,pos.y); lanes 16-31 hold K=2,3 (pos.z,1).
// B (4x16 f32) tile t: lanes 0-15 hold rows K=0,1 of W1aug; lanes 16-31 rows K=2,3;
//                      column N = lane & 15 (global col = 16t + N). Row 3 of W1aug = b1.
// D (16x16 f32): lane L, vgpr v -> row M = v + 8*(L>=16), col N = L&15.
__global__ __launch_bounds__(128)
void nerf_sigma_pass(const float* __restrict__ rays_o,
                     const float* __restrict__ rays_d,
                     const float* __restrict__ W1,   // [3,64] row-major
                     const float* __restrict__ b1,   // [64]
                     const float* __restrict__ W2,   // [64,1]
                     const float* __restrict__ b2,   // [1]
                     const float* __restrict__ t_starts,
                     const float* __restrict__ t_ends,
                     const int*   __restrict__ ray_indices,
                     float*       __restrict__ sigma_ws,
                     int S)
{
    const int lane = threadIdx.x & 31;
    const int wave = (blockIdx.x * (blockDim.x >> 5)) + (threadIdx.x >> 5);
    const int nTiles = (S + 15) >> 4;
    if (wave >= nTiles) return;            // uniform per wave: EXEC stays all-ones

    const int half = lane >> 4;            // 0 = lanes 0-15, 1 = lanes 16-31
    const int c    = lane & 15;            // A row / B column index within tile

    int si  = wave * 16 + c;               // this lane's sample (row of A)
    int sil = si < S ? si : S - 1;         // clamped index for loads

    // sigma_fn position at segment midpoint
    int   ri = ray_indices[sil];
    float ts = t_starts[sil];
    float te = t_ends[sil];
    float tm = 0.5f * (ts + te);
    float px = rays_o[ri * 3 + 0] + rays_d[ri * 3 + 0] * tm;
    float py = rays_o[ri * 3 + 1] + rays_d[ri * 3 + 1] * tm;
    float pz = rays_o[ri * 3 + 2] + rays_d[ri * 3 + 2] * tm;

    v2f A;
    A.x = half ? pz   : px;
    A.y = half ? 1.0f : py;

    v8f part = {};   // layer-2 partial sums: part[v] belongs to row M = v + 8*half

#if __has_builtin(__builtin_amdgcn_wmma_f32_16x16x4_f32)
    #pragma unroll
    for (int t = 0; t < 4; ++t) {
        const int col = t * 16 + c;
        v2f B;
        B.x = half ? W1[2 * NERF_HIDDEN + col] : W1[0 * NERF_HIDDEN + col];
        B.y = half ? b1[col]                   : W1[1 * NERF_HIDDEN + col];
        v8f Cz = {};
        v8f H = __builtin_amdgcn_wmma_f32_16x16x4_f32(
            /*neg_a=*/false, A, /*neg_b=*/false, B,
            /*c_mod=*/(short)0, Cz, /*reuse_a=*/false, /*reuse_b=*/false);
        const float w2 = W2[col];
        #pragma unroll
        for (int v = 0; v < 8; ++v) {
            float h = H[v] > 0.0f ? H[v] : 0.0f;   // ReLU
            part[v] += h * w2;
        }
    }
#else
    // VALU fallback (same data placement as the WMMA D layout).
    #pragma unroll
    for (int t = 0; t < 4; ++t) {
        const int col = t * 16 + c;
        const float w0 = W1[0 * NERF_HIDDEN + col];
        const float w1 = W1[1 * NERF_HIDDEN + col];
        const float wz = W1[2 * NERF_HIDDEN + col];
        const float bb = b1[col];
        const float w2 = W2[col];
        #pragma unroll
        for (int v = 0; v < 8; ++v) {
            const int m = 8 * half + v;            // row of the tile
            float qx = __shfl(px, m, 32);
            float qy = __shfl(py, m, 32);
            float qz = __shfl(pz, m, 32);
            float h = qx * w0 + qy * w1 + qz * wz + bb;
            h = h > 0.0f ? h : 0.0f;
            part[v] += h * w2;
        }
    }
#endif

    // Butterfly reduction within each 16-lane half: masks 1,2,4,8 never cross halves.
    #pragma unroll
    for (int m = 1; m <= 8; m <<= 1) {
        #pragma unroll
        for (int v = 0; v < 8; ++v)
            part[v] += __shfl_xor(part[v], m, 32);
    }
    // Now lanes 0-15 all hold rows 0..7 in part[0..7]; lanes 16-31 hold rows 8..15.

    const float bb2 = b2[0];
    float sg[8];
    #pragma unroll
    for (int v = 0; v < 8; ++v)
        sg[v] = softplus_stable(part[v] + bb2);

    if (c == 0) {                       // lanes 0 and 16 write 8 sigmas each
        const int base = wave * 16 + 8 * half;
        if (base + 7 < S) {
            float4 lo4 = make_float4(sg[0], sg[1], sg[2], sg[3]);
            float4 hi4 = make_float4(sg[4], sg[5], sg[6], sg[7]);
            *(float4*)(sigma_ws + base)     = lo4;
            *(float4*)(sigma_ws + base + 4) = hi4;
        } else {
            #pragma unroll
            for (int v = 0; v < 8; ++v)
                if (base + v < S) sigma_ws[base + v] = sg[v];
        }
    }
}

// One thread per ray: segmented exclusive cumsum -> weights -> opacity/depth.
__global__ __launch_bounds__(256)
void nerf_render_pass(const float* __restrict__ t_starts,
                      const float* __restrict__ t_ends,
                      const int*   __restrict__ ray_indices,
                      const float* __restrict__ sigma_ws,
                      float*       __restrict__ out,   // [n_rays, 2]
                      int S, int n_rays)
{
    const int r = blockIdx.x * blockDim.x + threadIdx.x;
    if (r >= n_rays) return;

    // lower_bound over sorted ray_indices (== jnp.searchsorted)
    int lo = 0, hi = S;
    while (lo < hi) { int mid = (lo + hi) >> 1; if (ray_indices[mid] < r)     lo = mid + 1; else hi = mid; }
    const int seg_lo = lo;
    lo = seg_lo; hi = S;
    while (lo < hi) { int mid = (lo + hi) >> 1; if (ray_indices[mid] < r + 1) lo = mid + 1; else hi = mid; }
    const int seg_hi = lo;

    float cum = 0.0f, opac = 0.0f, dnum = 0.0f;
    for (int i = seg_lo; i < seg_hi; ++i) {
        const float ts = t_starts[i];
        const float te = t_ends[i];
        const float dt = te - ts;
        const float tm = 0.5f * (ts + te);
        const float s  = sigma_ws[i] * dt;
        const float trans = expf(-cum);          // exp(-(excl - offset))
        const float alpha = 1.0f - expf(-s);
        const float w = trans * alpha;
        opac += w;
        dnum += w * tm;
        cum  += s;
    }
    const float depth = dnum / fmaxf(opac, 1.17549435e-38f);  // clip at f32 tiny
    out[2 * r + 0] = opac;
    out[2 * r + 1] = depth;
}

extern "C" void kernel_launch(void* const* d_in, const int* in_sizes, int n_in,
                              void* d_out, int out_size, void* d_ws, size_t ws_size,
                              hipStream_t stream) {
    const float* rays_o      = (const float*)d_in[0];
    const float* rays_d      = (const float*)d_in[1];
    const float* W1          = (const float*)d_in[2];
    const float* b1          = (const float*)d_in[3];
    const float* W2          = (const float*)d_in[4];
    const float* b2          = (const float*)d_in[5];
    const float* t_starts    = (const float*)d_in[6];
    const float* t_ends      = (const float*)d_in[7];
    const int*   ray_indices = (const int*)d_in[8];

    const int n_rays = in_sizes[0] / 3;   // rays_o is [n_rays, 3]
    const int S      = in_sizes[6];       // t_starts is [S]

    float* sigma_ws = (float*)d_ws;       // S floats of scratch

    const int nTiles = (S + 15) / 16;
    const int wavesPerBlock = 4;          // 128 threads
    const int blocks1 = (nTiles + wavesPerBlock - 1) / wavesPerBlock;
    nerf_sigma_pass<<<blocks1, 128, 0, stream>>>(
        rays_o, rays_d, W1, b1, W2, b2, t_starts, t_ends, ray_indices, sigma_ws, S);

    const int blocks2 = (n_rays + 255) / 256;
    nerf_render_pass<<<blocks2, 256, 0, stream>>>(
        t_starts, t_ends, ray_indices, sigma_ws, (float*)d_out, S, n_rays);
}